// SoftFeatureGouraudShader_70686571758251
// MI455X (gfx1250) — compile-verified
//
#include <hip/hip_runtime.h>
#include <cstdint>

// Reference constants
#define SIGMA_INV   1.0e4f          // 1/SIGMA
#define GAMMA_INV   1.0e4f          // 1/GAMMA
#define ZFAR_C      100.0f
#define ZRANGE_INV  0.01f           // 1/(ZFAR-ZNEAR)
#define DELTA_C     1.0000010000005e-10f  // exp(1e-10/GAMMA)*1e-10

constexpr int NPIX          = 8 * 128 * 128;   // N*H*W
constexpr int CCH           = 32;              // channels == wave32 lanes
constexpr int PIX_PER_BLOCK = 64;
constexpr int THREADS       = 256;             // 8 wave32s
constexpr int PIX_PER_WAVE  = PIX_PER_BLOCK / (THREADS / 32);

// gfx1250 async global->LDS copy (ASYNCcnt path). vdst = LDS byte address,
// vaddr = 64-bit global address, per-lane, EXEC-masked.
__device__ __forceinline__ void async_b128(uint32_t lds_byte, uint64_t gaddr) {
    asm volatile("global_load_async_to_lds_b128 %0, %1, off"
                 :: "v"(lds_byte), "v"(gaddr)
                 : "memory");
}

__global__ __launch_bounds__(THREADS)
void SoftFeatureGouraudShader_kernel(const float* __restrict__ feature,  // (V,32)
                                     const int*   __restrict__ faces,    // (F,3)
                                     const int*   __restrict__ p2f,      // (P,4)
                                     const float* __restrict__ bary,     // (P,4,3)
                                     const float* __restrict__ dists,    // (P,4)
                                     const float* __restrict__ zbuf,     // (P,4)
                                     float*       __restrict__ out)      // (P,33)
{
    // LDS staging: bary[64*12f]=3072B | p2f[64*4i]=1024B | dists=1024B | zbuf=1024B
    __shared__ __align__(16) uint8_t smem[6144];
    const uint32_t lds_base = (uint32_t)(uintptr_t)(&smem[0]); // low 32 bits = LDS offset

    const int tid  = threadIdx.x;
    const int base = blockIdx.x * PIX_PER_BLOCK;

    // ---- async stage of per-pixel blend inputs (<=2 ops/lane) ----
    if (tid < 192) {   // bary: 192 x 16B
        async_b128(lds_base + (uint32_t)tid * 16u,
                   (uint64_t)(uintptr_t)(bary + (size_t)base * 12 + (size_t)tid * 4));
    } else {           // pix_to_face: 64 x 16B
        const int i = tid - 192;
        async_b128(lds_base + 3072u + (uint32_t)i * 16u,
                   (uint64_t)(uintptr_t)(p2f + (size_t)base * 4 + (size_t)i * 4));
    }
    if (tid < 64) {    // dists: 64 x 16B
        async_b128(lds_base + 4096u + (uint32_t)tid * 16u,
                   (uint64_t)(uintptr_t)(dists + (size_t)base * 4 + (size_t)tid * 4));
    } else if (tid < 128) {  // zbuf: 64 x 16B
        const int i = tid - 64;
        async_b128(lds_base + 5120u + (uint32_t)i * 16u,
                   (uint64_t)(uintptr_t)(zbuf + (size_t)base * 4 + (size_t)i * 4));
    }
    asm volatile("s_wait_asynccnt 0x0" ::: "memory");
    __syncthreads();

    const float* s_bary  = (const float*)(smem);
    const int*   s_p2f   = (const int*)  (smem + 3072);
    const float* s_dists = (const float*)(smem + 4096);
    const float* s_zbuf  = (const float*)(smem + 5120);

    const int wave = tid >> 5;
    const int lane = tid & 31;          // lane == channel (wave32 == C)

    for (int it = 0; it < PIX_PER_WAVE; ++it) {
        const int lp = wave * PIX_PER_WAVE + it;   // local pixel
        const int p  = base + lp;                  // global pixel

        // wave-uniform per-pixel data via LDS broadcast reads
        int   fi[4];
        float br[12], dd[4], zz[4];
        {
            const int4   f4 = ((const int4*)  s_p2f)  [lp];
            const float4 d4 = ((const float4*)s_dists)[lp];
            const float4 z4 = ((const float4*)s_zbuf) [lp];
            fi[0]=f4.x; fi[1]=f4.y; fi[2]=f4.z; fi[3]=f4.w;
            dd[0]=d4.x; dd[1]=d4.y; dd[2]=d4.z; dd[3]=d4.w;
            zz[0]=z4.x; zz[1]=z4.y; zz[2]=z4.z; zz[3]=z4.w;
            const float4* bv = (const float4*)s_bary + lp * 3;
            *(float4*)(br + 0) = bv[0];
            *(float4*)(br + 4) = bv[1];
            *(float4*)(br + 8) = bv[2];
        }

        // ---- scalar face-vertex fetches (SMEM/KMcnt), branchless: clamp to 0 ----
        int vid[12];
#pragma unroll
        for (int k = 0; k < 4; ++k) {
            const int fs = __builtin_amdgcn_readfirstlane(fi[k] < 0 ? 0 : fi[k]);
            vid[k * 3 + 0] = faces[3 * fs + 0];
            vid[k * 3 + 1] = faces[3 * fs + 1];
            vid[k * 3 + 2] = faces[3 * fs + 2];
        }

        // ---- issue ALL 12 feature gathers before any consumption ----
        float fv[12];
#pragma unroll
        for (int j = 0; j < 12; ++j)
            fv[j] = feature[(size_t)vid[j] * CCH + lane];

        // ---- soft blend weights, overlapped with gather latency ----
        float prob[4], zinv[4];
        float alpha = 1.0f;
#pragma unroll
        for (int k = 0; k < 4; ++k) {
            const bool m = fi[k] >= 0;
            const float pr = m ? __builtin_amdgcn_rcpf(1.0f + expf(dd[k] * SIGMA_INV)) : 0.0f;
            const float zi = m ? (ZFAR_C - zz[k]) * ZRANGE_INV : 0.0f;
            prob[k] = pr; zinv[k] = zi;
            alpha *= (1.0f - pr);
        }
        const float zmax = fmaxf(fmaxf(zinv[0], zinv[1]), fmaxf(zinv[2], zinv[3]));
        float num[4];
        float denom = DELTA_C;
#pragma unroll
        for (int k = 0; k < 4; ++k) {
            num[k] = prob[k] * expf((zinv[k] - zmax) * GAMMA_INV);
            denom += num[k];
        }
        const float inv_denom = __builtin_amdgcn_rcpf(denom);

        // ---- consume gathers: masked slots have wk == 0 ----
        float acc = 0.0f;
#pragma unroll
        for (int k = 0; k < 4; ++k) {
            const float wk = num[k] * inv_denom;
            acc = fmaf(br[k * 3 + 0] * wk, fv[k * 3 + 0],
                  fmaf(br[k * 3 + 1] * wk, fv[k * 3 + 1],
                  fmaf(br[k * 3 + 2] * wk, fv[k * 3 + 2], acc)));
        }

        // output: 32 rgb channels (coalesced) + alpha from lane 0
        __builtin_nontemporal_store(acc, out + (size_t)p * 33 + lane);
        if (lane == 0)
            __builtin_nontemporal_store(1.0f - alpha, out + (size_t)p * 33 + 32);
    }
}

extern "C" void kernel_launch(void* const* d_in, const int* in_sizes, int n_in,
                              void* d_out, int out_size, void* d_ws, size_t ws_size,
                              hipStream_t stream) {
    const float* feature = (const float*)d_in[0];
    const int*   faces   = (const int*)  d_in[1];
    const int*   p2f     = (const int*)  d_in[2];
    const float* bary    = (const float*)d_in[3];
    const float* dists   = (const float*)d_in[4];
    const float* zbuf    = (const float*)d_in[5];
    float* out = (float*)d_out;

    dim3 grid(NPIX / PIX_PER_BLOCK);   // 2048 blocks
    dim3 block(THREADS);               // 8 wave32s
    SoftFeatureGouraudShader_kernel<<<grid, block, 0, stream>>>(
        feature, faces, p2f, bary, dists, zbuf, out);
}